// BiMultiHeadAttention_69234872811760
// MI455X (gfx1250) — compile-verified
//
#include <hip/hip_runtime.h>
#include <cstddef>

// ---------------- problem constants (static per reference) ----------------
#define B_      2
#define S_      128
#define T_      8525            // 80*80+40*40+20*20+10*10+5*5
#define BT_     (B_*T_)         // 17050
#define BS_     (B_*S_)         // 256
#define HEADS_  8
#define HD_     32
#define EMBED_  256
#define VDIM_   256
#define LDIM_   768
#define BH_     (B_*HEADS_)     // 16
#define SCALEQ  0.1767766952966369f   // 32^-0.5
#define CLAMP_  50000.0f
#define NEGV_   (-9.0e15f)

typedef __attribute__((ext_vector_type(2))) float v2f;
typedef __attribute__((ext_vector_type(4))) float f4v;
typedef __attribute__((ext_vector_type(8))) float v8f;

// D = A(16x4) * B(4x16) + C(16x16), exact fp32 tensor op on CDNA5
__device__ __forceinline__ v8f wmma4(v2f a, v2f b, v8f c) {
  return __builtin_amdgcn_wmma_f32_16x16x4_f32(false, a, false, b, (short)0, c,
                                               false, false);
}
__device__ __forceinline__ float clampv(float x) {
  return fminf(fmaxf(x, -CLAMP_), CLAMP_);
}
// monotonic uint encoding of float for atomic max over signed values
__device__ __forceinline__ unsigned encf(float x) {
  unsigned u = __float_as_uint(x);
  return (u & 0x80000000u) ? ~u : (u | 0x80000000u);
}
__device__ __forceinline__ float decf(unsigned u) {
  return __uint_as_float((u & 0x80000000u) ? (u & 0x7fffffffu) : ~u);
}

__device__ __forceinline__ float block_reduce_max(float v, float* red, int tid) {
  red[tid] = v; __syncthreads();
  for (int o = 128; o > 0; o >>= 1) {
    if (tid < o) red[tid] = fmaxf(red[tid], red[tid + o]);
    __syncthreads();
  }
  float r = red[0]; __syncthreads(); return r;
}
__device__ __forceinline__ float block_reduce_sum(float v, float* red, int tid) {
  red[tid] = v; __syncthreads();
  for (int o = 128; o > 0; o >>= 1) {
    if (tid < o) red[tid] += red[tid + o];
    __syncthreads();
  }
  float r = red[0]; __syncthreads(); return r;
}

// ---------------- init ----------------
__global__ void init_kernel(unsigned* gmax) {
  if (threadIdx.x == 0 && blockIdx.x == 0) *gmax = 0u;   // < encf(-inf)
}

// ---------------- generic WMMA GEMM:  Y = (X[MxK] @ W[KxN] + bias[N]) * scale
// block = 256 threads = 8 waves; block tile 64(M) x 64(N); K multiple of 64.
__global__ __launch_bounds__(256) void gemm_bias(
    const float* __restrict__ X, const float* __restrict__ W,
    const float* __restrict__ bias, float* __restrict__ Y,
    int M, int N, int K, float scale)
{
  __shared__ float As[64][68];   // [m][k]   (68: 16B-aligned rows, conflict pad)
  __shared__ float Bs[64][68];   // [n][k]   (B transposed for b64 reads)
  const int tid = threadIdx.x;
  const int lane = tid & 31;
  const int w  = tid >> 5;
  const int wm = w & 3;          // 16-row group within 64
  const int wn = w >> 2;         // 32-col group within 64
  const int lo = lane & 15;
  const int hi = lane >> 4;
  const int m0 = blockIdx.y * 64;
  const int n0 = blockIdx.x * 64;

  v8f acc0 = {}; v8f acc1 = {};

  for (int kc = 0; kc < K; kc += 64) {
    // A tile 64x64, float4 coalesced
#pragma unroll
    for (int i = 0; i < 4; ++i) {
      int idx = tid + i * 256;          // vec4 index 0..1023
      int row = idx >> 4;
      int kq  = (idx & 15) << 2;
      f4v val = {};
      if (m0 + row < M)
        val = *(const f4v*)&X[(size_t)(m0 + row) * K + kc + kq];
      *(f4v*)&As[row][kq] = val;
    }
    // B tile: W[kc+kr][n0+nc] -> Bs[nc][kr], coalesced over nc
#pragma unroll
    for (int i = 0; i < 16; ++i) {
      int idx = tid + i * 256;          // 0..4095
      int kr  = idx >> 6;
      int nc  = idx & 63;
      Bs[nc][kr] = W[(size_t)(kc + kr) * N + n0 + nc];
    }
    __syncthreads();
#pragma unroll
    for (int kk = 0; kk < 64; kk += 4) {
      v2f a  = *(const v2f*)&As[wm * 16 + lo][kk + 2 * hi];
      v2f b0 = *(const v2f*)&Bs[wn * 32 + lo][kk + 2 * hi];
      v2f b1 = *(const v2f*)&Bs[wn * 32 + 16 + lo][kk + 2 * hi];
      acc0 = wmma4(a, b0, acc0);
      acc1 = wmma4(a, b1, acc1);
    }
    __syncthreads();
  }
#pragma unroll
  for (int r = 0; r < 8; ++r) {
    int row = m0 + wm * 16 + r + 8 * hi;
    if (row < M) {
      int c0 = n0 + wn * 32 + lo;
      int c1 = c0 + 16;
      Y[(size_t)row * N + c0] = (acc0[r] + bias[c0]) * scale;
      Y[(size_t)row * N + c1] = (acc1[r] + bias[c1]) * scale;
    }
  }
}

// ---------------- aw[bh][t][s] = sum_d q[b,t,h,d] * k[b,s,h,d]  (K=32)
// also accumulates global max of raw scores into *gmax (monotonic-uint).
__global__ __launch_bounds__(256) void aw_kernel(
    const float* __restrict__ q, const float* __restrict__ kmat,
    float* __restrict__ aw, unsigned* __restrict__ gmax)
{
  __shared__ float As[64][36];    // q tile [t][d]
  __shared__ float Bs[128][36];   // k      [s][d]
  __shared__ unsigned smax;
  const int tid = threadIdx.x;
  const int lane = tid & 31;
  const int w  = tid >> 5;
  const int wm = w & 3;           // 4 x 16 = 64 t-rows
  const int wn = w >> 2;          // 2 x 64 = 128 s-cols
  const int lo = lane & 15;
  const int hi = lane >> 4;
  const int bh = blockIdx.y;
  const int b = bh >> 3, h = bh & 7;
  const int t0 = blockIdx.x * 64;
  if (tid == 0) smax = 0u;

#pragma unroll
  for (int i = 0; i < 2; ++i) {                 // 512 vec4 = 64x32
    int idx = tid + i * 256;
    int row = idx >> 3;
    int d4  = (idx & 7) << 2;
    f4v val = {};
    if (t0 + row < T_)
      val = *(const f4v*)&q[(size_t)(b * T_ + t0 + row) * EMBED_ + h * HD_ + d4];
    *(f4v*)&As[row][d4] = val;
  }
#pragma unroll
  for (int i = 0; i < 4; ++i) {                 // 1024 vec4 = 128x32
    int idx = tid + i * 256;
    int row = idx >> 3;
    int d4  = (idx & 7) << 2;
    *(f4v*)&Bs[row][d4] =
        *(const f4v*)&kmat[(size_t)(b * S_ + row) * EMBED_ + h * HD_ + d4];
  }
  __syncthreads();

  v8f acc[4] = {};
#pragma unroll
  for (int kk = 0; kk < 32; kk += 4) {
    v2f a = *(const v2f*)&As[wm * 16 + lo][kk + 2 * hi];
#pragma unroll
    for (int nt = 0; nt < 4; ++nt) {
      v2f bb = *(const v2f*)&Bs[wn * 64 + nt * 16 + lo][kk + 2 * hi];
      acc[nt] = wmma4(a, bb, acc[nt]);
    }
  }

  float lmax = -INFINITY;
#pragma unroll
  for (int nt = 0; nt < 4; ++nt) {
#pragma unroll
    for (int r = 0; r < 8; ++r) {
      int t = t0 + wm * 16 + r + 8 * hi;
      int s = wn * 64 + nt * 16 + lo;
      if (t < T_) {
        float vv = acc[nt][r];
        aw[((size_t)bh * T_ + t) * S_ + s] = vv;
        lmax = fmaxf(lmax, vv);
      }
    }
  }
  atomicMax(&smax, encf(lmax));
  __syncthreads();
  if (tid == 0) atomicMax(gmax, smax);
}

// ---------------- per-(bh,s) column stats for the double softmax over t
// stats = { max1, 1/sum1, 1/sum2 }
__global__ __launch_bounds__(256) void lstats_kernel(
    const float* __restrict__ aw, const unsigned* __restrict__ gmaxu,
    float* __restrict__ stats)
{
  __shared__ float red[256];
  const int tid = threadIdx.x;
  const int s  = blockIdx.x;
  const int bh = blockIdx.y;
  const float gmax = decf(*gmaxu);
  const float* col = aw + (size_t)bh * T_ * S_ + s;

  float m = -INFINITY;
  for (int t = tid; t < T_; t += 256) {
    if (t + 2048 < T_) __builtin_prefetch(&col[(size_t)(t + 2048) * S_], 0, 1);
    m = fmaxf(m, clampv(col[(size_t)t * S_] - gmax));
  }
  float max1 = block_reduce_max(m, red, tid);

  float sm = 0.f;
  for (int t = tid; t < T_; t += 256)
    sm += __expf(clampv(col[(size_t)t * S_] - gmax) - max1);
  float sum1 = block_reduce_sum(sm, red, tid);
  float inv1 = 1.0f / sum1;                    // also == max of awl (= exp(0)/sum1)

  float s2 = 0.f;
  for (int t = tid; t < T_; t += 256) {
    float awl = __expf(clampv(col[(size_t)t * S_] - gmax) - max1) * inv1;
    s2 += __expf(awl - inv1);
  }
  float sum2 = block_reduce_sum(s2, red, tid);

  if (tid == 0) {
    float* st = &stats[(size_t)(bh * S_ + s) * 3];
    st[0] = max1; st[1] = inv1; st[2] = 1.0f / sum2;
  }
}

// ---------------- out_l heads: [128 s] x [T] double-softmax weights @ val_v[T x 32]
// one block per bh; 8 waves, each 16 s-rows x 32 d-cols; K looped in 64-chunks.
__global__ __launch_bounds__(256) void outl_kernel(
    const float* __restrict__ aw, const float* __restrict__ valv,
    const float* __restrict__ stats, const unsigned* __restrict__ gmaxu,
    float* __restrict__ outlh)
{
  __shared__ float As[128][68];   // weights [s][t-chunk]
  __shared__ float Bs[32][68];    // val_v   [d][t-chunk]
  __shared__ float st[128][4];    // max1, inv1, inv2
  const int tid = threadIdx.x;
  const int lane = tid & 31;
  const int w  = tid >> 5;        // wm: 8 x 16 = 128 s-rows
  const int lo = lane & 15;
  const int hi = lane >> 4;
  const int bh = blockIdx.x;
  const int b = bh >> 3, h = bh & 7;
  const float gmax = decf(*gmaxu);

  if (tid < 128) {
    const float* p = &stats[(size_t)(bh * S_ + tid) * 3];
    st[tid][0] = p[0]; st[tid][1] = p[1]; st[tid][2] = p[2];
  }
  __syncthreads();

  v8f acc0 = {}; v8f acc1 = {};
  for (int tc = 0; tc < T_; tc += 64) {
    // fused weight transform: aw[t][s] -> w[s][t]
#pragma unroll
    for (int i = 0; i < 32; ++i) {
      int idx = tid + i * 256;       // 0..8191
      int tt = idx >> 7;
      int s  = idx & 127;
      int t  = tc + tt;
      float wv = 0.f;
      if (t < T_) {
        float val = clampv(aw[((size_t)bh * T_ + t) * S_ + s] - gmax);
        float awl = __expf(val - st[s][0]) * st[s][1];
        wv = __expf(awl - st[s][1]) * st[s][2];
      }
      As[s][tt] = wv;
    }
#pragma unroll
    for (int i = 0; i < 8; ++i) {
      int idx = tid + i * 256;       // 0..2047
      int tt = idx >> 5;
      int d  = idx & 31;
      int t  = tc + tt;
      Bs[d][tt] = (t < T_)
          ? valv[(size_t)(b * T_ + t) * EMBED_ + h * HD_ + d] : 0.f;
    }
    __syncthreads();
#pragma unroll
    for (int kk = 0; kk < 64; kk += 4) {
      v2f a  = *(const v2f*)&As[w * 16 + lo][kk + 2 * hi];
      v2f b0 = *(const v2f*)&Bs[lo][kk + 2 * hi];
      v2f b1 = *(const v2f*)&Bs[16 + lo][kk + 2 * hi];
      acc0 = wmma4(a, b0, acc0);
      acc1 = wmma4(a, b1, acc1);
    }
    __syncthreads();
  }
#pragma unroll
  for (int r = 0; r < 8; ++r) {
    int s = w * 16 + r + 8 * hi;
    outlh[(size_t)(b * S_ + s) * EMBED_ + h * HD_ + lo]      = acc0[r];
    outlh[(size_t)(b * S_ + s) * EMBED_ + h * HD_ + 16 + lo] = acc1[r];
  }
}

// ---------------- out_v heads: row softmax(aw + mask) @ val_l  (K = 128)
__global__ __launch_bounds__(256) void outv_kernel(
    const float* __restrict__ aw, const float* __restrict__ vall,
    const float* __restrict__ maskl, const unsigned* __restrict__ gmaxu,
    float* __restrict__ outvh)
{
  __shared__ float P[64][132];    // scores -> probs, [t][s]
  __shared__ float Bs[32][132];   // val_l [d][s]
  __shared__ float addm[128];
  __shared__ float rmax[64], rsum[64];
  const int tid = threadIdx.x;
  const int lane = tid & 31;
  const int w  = tid >> 5;
  const int wm = w & 3;           // 4 x 16 = 64 t-rows
  const int wn = w >> 2;          // 2 x 16 = 32 d-cols
  const int lo = lane & 15;
  const int hi = lane >> 4;
  const int bh = blockIdx.y;
  const int b = bh >> 3, h = bh & 7;
  const int t0 = blockIdx.x * 64;
  const float gmax = decf(*gmaxu);

  if (tid < 128) {
    float mk = maskl[b * S_ + tid];
    addm[tid] = (mk == 0.f) ? NEGV_ : mk;
  }
#pragma unroll
  for (int i = 0; i < 16; ++i) {   // val_l: 128x32 -> Bs[d][s]
    int idx = tid + i * 256;       // 0..4095
    int s = idx >> 5;
    int d = idx & 31;
    Bs[d][s] = vall[(size_t)(b * S_ + s) * EMBED_ + h * HD_ + d];
  }
  __syncthreads();
#pragma unroll
  for (int i = 0; i < 32; ++i) {
    int idx = tid + i * 256;       // 0..8191
    int tt = idx >> 7;
    int s  = idx & 127;
    int t  = t0 + tt;
    P[tt][s] = (t < T_)
        ? clampv(aw[((size_t)bh * T_ + t) * S_ + s] - gmax) + addm[s]
        : NEGV_;
  }
  __syncthreads();
  if (tid < 64) {
    float m = -INFINITY;
    for (int s = 0; s < S_; ++s) m = fmaxf(m, P[tid][s]);
    float sum = 0.f;
    for (int s = 0; s < S_; ++s) sum += __expf(P[tid][s] - m);
    rmax[tid] = m; rsum[tid] = 1.0f / sum;
  }
  __syncthreads();
#pragma unroll
  for (int i = 0; i < 32; ++i) {
    int idx = tid + i * 256;
    int tt = idx >> 7;
    int s  = idx & 127;
    P[tt][s] = __expf(P[tt][s] - rmax[tt]) * rsum[tt];
  }
  __syncthreads();

  v8f acc = {};
#pragma unroll
  for (int kk = 0; kk < 128; kk += 4) {
    v2f a  = *(const v2f*)&P[wm * 16 + lo][kk + 2 * hi];
    v2f bb = *(const v2f*)&Bs[wn * 16 + lo][kk + 2 * hi];
    acc = wmma4(a, bb, acc);
  }
#pragma unroll
  for (int r = 0; r < 8; ++r) {
    int t = t0 + wm * 16 + r + 8 * hi;
    if (t < T_) {
      int d = wn * 16 + lo;
      outvh[(size_t)(b * T_ + t) * EMBED_ + h * HD_ + d] = acc[r];
    }
  }
}

// ---------------- host orchestration ----------------
extern "C" void kernel_launch(void* const* d_in, const int* in_sizes, int n_in,
                              void* d_out, int out_size, void* d_ws, size_t ws_size,
                              hipStream_t stream)
{
  (void)in_sizes; (void)n_in; (void)out_size; (void)ws_size;
  const float* v    = (const float*)d_in[0];
  const float* l    = (const float*)d_in[1];
  const float* mask = (const float*)d_in[2];
  // d_in[3] = size_per_level (static, unused)
  const float* vpw  = (const float*)d_in[4];
  const float* vpb  = (const float*)d_in[5];
  const float* lpw  = (const float*)d_in[6];
  const float* lpb  = (const float*)d_in[7];
  const float* vvpw = (const float*)d_in[8];
  const float* vvpb = (const float*)d_in[9];
  const float* vlpw = (const float*)d_in[10];
  const float* vlpb = (const float*)d_in[11];
  const float* ovpw = (const float*)d_in[12];
  const float* ovpb = (const float*)d_in[13];
  const float* olpw = (const float*)d_in[14];
  const float* olpb = (const float*)d_in[15];
  float* out = (float*)d_out;

  float* ws = (float*)d_ws;
  size_t off = 0;
  float* q     = ws + off; off += (size_t)BT_ * EMBED_;   // 4,364,800
  float* valv  = ws + off; off += (size_t)BT_ * EMBED_;
  float* kbuf  = ws + off; off += (size_t)BS_ * EMBED_;
  float* vall  = ws + off; off += (size_t)BS_ * EMBED_;
  float* aw    = ws + off; off += (size_t)BH_ * T_ * S_;  // 17,459,200
  float* outvh = ws + off; off += (size_t)BT_ * EMBED_;
  float* outlh = ws + off; off += (size_t)BS_ * EMBED_;
  float* stats = ws + off; off += (size_t)BH_ * S_ * 3;
  unsigned* gmax = (unsigned*)(ws + off);                 // ~123 MB total

  const dim3 blk(256);
  const int mtiles = (BT_ + 63) / 64;   // 267
  const int ttiles = (T_ + 63) / 64;    // 134

  init_kernel<<<1, 32, 0, stream>>>(gmax);

  // projections
  gemm_bias<<<dim3(EMBED_ / 64, mtiles), blk, 0, stream>>>(v, vpw,  vpb,  q,    BT_, EMBED_, VDIM_, SCALEQ);
  gemm_bias<<<dim3(EMBED_ / 64, mtiles), blk, 0, stream>>>(v, vvpw, vvpb, valv, BT_, EMBED_, VDIM_, 1.0f);
  gemm_bias<<<dim3(EMBED_ / 64, (BS_ + 63) / 64), blk, 0, stream>>>(l, lpw,  lpb,  kbuf, BS_, EMBED_, LDIM_, 1.0f);
  gemm_bias<<<dim3(EMBED_ / 64, (BS_ + 63) / 64), blk, 0, stream>>>(l, vlpw, vlpb, vall, BS_, EMBED_, LDIM_, 1.0f);

  // attention scores + global max
  aw_kernel<<<dim3(ttiles, BH_), blk, 0, stream>>>(q, kbuf, aw, gmax);

  // column-softmax (double softmax) stats, then out_l heads
  lstats_kernel<<<dim3(S_, BH_), blk, 0, stream>>>(aw, gmax, stats);
  outl_kernel<<<dim3(BH_), blk, 0, stream>>>(aw, valv, stats, gmax, outlh);

  // row softmax + out_v heads
  outv_kernel<<<dim3(ttiles, BH_), blk, 0, stream>>>(aw, vall, mask, gmax, outvh);

  // output projections (tuple order: out_v flat, then out_l)
  gemm_bias<<<dim3(VDIM_ / 64, mtiles), blk, 0, stream>>>(outvh, ovpw, ovpb, out, BT_, VDIM_, EMBED_, 1.0f);
  gemm_bias<<<dim3(LDIM_ / 64, (BS_ + 63) / 64), blk, 0, stream>>>(outlh, olpw, olpb, out + (size_t)BT_ * VDIM_, BS_, LDIM_, EMBED_, 1.0f);
}